// ViewpointLoss_70643622085336
// MI455X (gfx1250) — compile-verified
//
#include <hip/hip_runtime.h>
#include <hip/hip_bf16.h>

// ViewpointLoss for MI455X (gfx1250, wave32).
//   per row: l1 = sum|x|; inv = 1/l1; S = sum exp(x*inv); g = x[label]*inv - log(S)
//   loss = -sum_rows(g) / C      (C = 360)
//
// Design: one wave32 per row, row resident in 12 VGPRs (360 = 11*32 + 8).
// Single pass over HBM (~188.7 MB @ 23.3 TB/s => ~8.1 us floor).
// Wave-level 32-lane sums are done with V_WMMA_F32_16X16X4_F32 (A = lane
// partials in the documented 16x4 A layout, B = ones => D[m] = p(m)+p(m+16)),
// then 7 adds + one xor-16 swizzle broadcast the total to all lanes.

#define CP 360          // CLASS_PERIOD
#define REGS 12         // ceil(360/32)
#define FULLK 11        // 11*32 = 352 full chunks
#define TAIL 8          // 360 - 352 lanes participate in chunk 11

typedef __attribute__((ext_vector_type(2))) float v2f;
typedef __attribute__((ext_vector_type(8))) float v8f;

// Sum one float per lane across the full wave32 using WMMA f32 16x16x4.
// A[m][0] = partial(lane m), A[m][2] = partial(lane m+16), A[m][1]=A[m][3]=0,
// B = ones  =>  D[m][n] = partial(m) + partial(m+16) for every column n.
// Lane n (<16) holds D rows 0..7 in c[0..7]; lane n (>=16) holds rows 8..15.
// Sum the 8 accumulator VGPRs, then swap halves with xor-16 to get the total.
// NOTE: must be called with all 32 lanes active (EXEC = all ones).
__device__ __forceinline__ float wave_sum_wmma(float partial) {
    v2f a; a[0] = partial; a[1] = 0.0f;
    v2f b; b[0] = 1.0f;    b[1] = 1.0f;
    v8f c = {};
    c = __builtin_amdgcn_wmma_f32_16x16x4_f32(
        /*neg_a=*/false, a, /*neg_b=*/false, b,
        /*c_mod=*/(short)0, c, /*reuse_a=*/false, /*reuse_b=*/false);
    float t = ((c[0] + c[1]) + (c[2] + c[3])) + ((c[4] + c[5]) + (c[6] + c[7]));
    t += __shfl_xor(t, 16, 32);   // combine rows 0-7 half with rows 8-15 half
    return t;
}

__global__ void __launch_bounds__(256)
viewpoint_loss_stage1(const float* __restrict__ preds,
                      const int* __restrict__ labels,
                      float* __restrict__ block_partials,
                      int rows) {
    const int lane    = threadIdx.x & 31;
    const int waveIn  = threadIdx.x >> 5;           // 0..7
    const int wavesPerBlk = blockDim.x >> 5;        // 8
    const int gwave   = blockIdx.x * wavesPerBlk + waveIn;
    const int nwaves  = gridDim.x * wavesPerBlk;

    float acc = 0.0f;   // meaningful in lane 0 only

    for (int row = gwave; row < rows; row += nwaves) {
        const float* p = preds + (size_t)row * CP;

        // Prefetch this wave's next row (one byte per line pulls whole line).
        int nrow = row + nwaves;
        if (nrow < rows && lane < 12) {
            __builtin_prefetch(preds + (size_t)nrow * CP + lane * 32, 0, 0);
        }

        // ---- load row into registers: x[k] = p[k*32 + lane] ----
        float x[REGS];
#pragma unroll
        for (int k = 0; k < FULLK; ++k) x[k] = p[k * 32 + lane];
        x[FULLK] = (lane < TAIL) ? p[FULLK * 32 + lane] : 0.0f;

        // ---- pass 1: l1 = sum |x| over the row (WMMA wave reduction) ----
        float s = 0.0f;
#pragma unroll
        for (int k = 0; k < REGS; ++k) s += fabsf(x[k]);   // |0| = 0 for tail
        float l1  = wave_sum_wmma(s);
        float inv = 1.0f / l1;

        // ---- pass 2: S = sum exp(x * inv) over the row ----
        float e = 0.0f;
#pragma unroll
        for (int k = 0; k < FULLK; ++k) e += __expf(x[k] * inv);
        e += (lane < TAIL) ? __expf(x[FULLK] * inv) : 0.0f;  // don't add exp(0)
        float S    = wave_sum_wmma(e);
        float logS = __logf(S);

        // ---- gather: logp[label] = x[label]*inv - log(S) ----
        if (lane == 0) {
            int   lab = labels[row];
            float g   = p[lab];          // just-read row: L1/L2 hit
            acc += g * inv - logS;
        }
    }

    // ---- block reduction: one partial per block ----
    __shared__ float wsum[8];
    if (lane == 0) wsum[waveIn] = acc;
    __syncthreads();
    if (threadIdx.x == 0) {
        float t = 0.0f;
#pragma unroll
        for (int i = 0; i < 8; ++i) t += wsum[i];
        block_partials[blockIdx.x] = t;
    }
}

__global__ void __launch_bounds__(256)
viewpoint_loss_stage2(const float* __restrict__ partials, int n,
                      float* __restrict__ out) {
    float t = 0.0f;
    for (int i = threadIdx.x; i < n; i += blockDim.x) t += partials[i];
#pragma unroll
    for (int m = 16; m >= 1; m >>= 1) t += __shfl_xor(t, m, 32);
    __shared__ float wsum[8];
    if ((threadIdx.x & 31) == 0) wsum[threadIdx.x >> 5] = t;
    __syncthreads();
    if (threadIdx.x == 0) {
        float tot = 0.0f;
#pragma unroll
        for (int i = 0; i < 8; ++i) tot += wsum[i];
        out[0] = -tot * (1.0f / (float)CP);
    }
}

extern "C" void kernel_launch(void* const* d_in, const int* in_sizes, int n_in,
                              void* d_out, int out_size, void* d_ws, size_t ws_size,
                              hipStream_t stream) {
    const float* preds  = (const float*)d_in[0];
    const int*   labels = (const int*)d_in[1];
    float*       outp   = (float*)d_out;
    float*       part   = (float*)d_ws;
    const int rows = in_sizes[1];           // 131072

    int grid = 1024;                        // 8192 waves -> 16 rows/wave
    size_t need = (size_t)grid * sizeof(float);
    if (ws_size < need) {                   // stay inside workspace if small
        grid = (int)(ws_size / sizeof(float));
        if (grid < 1) grid = 1;
    }

    viewpoint_loss_stage1<<<grid, 256, 0, stream>>>(preds, labels, part, rows);
    viewpoint_loss_stage2<<<1, 256, 0, stream>>>(part, grid, outp);
}